// RippleNetModel_80994493268130
// MI455X (gfx1250) — compile-verified
//
#include <hip/hip_runtime.h>

#define LEAKY 0.2f

typedef float f4  __attribute__((ext_vector_type(4)));
typedef float v2f __attribute__((ext_vector_type(2)));
typedef float v8f __attribute__((ext_vector_type(8)));

__device__ __forceinline__ float leaky1(float x) {
    return x > 0.f ? x : LEAKY * x;
}

// ---------------- Kernel: plain vectorized copy (uEmbed / iEmbed passthrough) ----
__global__ void copy_f4(const f4* __restrict__ src, f4* __restrict__ dst, int n4) {
    int i = blockIdx.x * blockDim.x + threadIdx.x;
    if (i < n4) {
        f4 v = src[i];
        __builtin_nontemporal_store(v, &dst[i]);
    }
}

// ---------------- Kernel: gather h_all and t_all (leaky relu on t) --------------
// thread i -> (slot = i>>3, j = i&7), each slot row = 32 floats = 8 x float4
__global__ void gather_ht(const int* __restrict__ mh, const int* __restrict__ mt,
                          const f4* __restrict__ ent4,
                          f4* __restrict__ out_h4, f4* __restrict__ out_t4,
                          int slots) {
    int i = blockIdx.x * blockDim.x + threadIdx.x;
    if (i >= slots * 8) return;
    int s = i >> 3, j = i & 7;
    int hIdx = mh[s];
    int tIdx = mt[s];
    f4 hv = ent4[(size_t)hIdx * 8 + j];
    __builtin_nontemporal_store(hv, &out_h4[i]);
    f4 tv = ent4[(size_t)tIdx * 8 + j];
    tv.x = leaky1(tv.x); tv.y = leaky1(tv.y);
    tv.z = leaky1(tv.z); tv.w = leaky1(tv.w);
    __builtin_nontemporal_store(tv, &out_t4[i]);
}

// ---------------- Kernel: gather r_all (537 MB stream; one block per slot) -------
// block s copies relation_emb[mr[s]] (1024 floats = 256 x float4), coalesced.
__global__ void gather_r(const int* __restrict__ mr, const f4* __restrict__ rel4,
                         f4* __restrict__ out_r4) {
    int s = blockIdx.x;
    int rIdx = mr[s];
    f4 v = rel4[(size_t)rIdx * 256 + threadIdx.x];
    __builtin_nontemporal_store(v, &out_r4[(size_t)s * 256 + threadIdx.x]);
}

// ---------------- Kernel: 2-hop ripple key-addressing ---------------------------
// 16 waves / block; wave w owns batch element b = blockIdx*16 + w.
// Per hop: per-slot 32x32 matvec (R@h) with h broadcast via shfl, wave-wide
// softmax, weighted tail sum, then wave 0 applies X <- X @ W^T for all 16 rows
// with V_WMMA_F32_16X16X4_F32 (fp32 WMMA, matches reference precision).
__global__ __launch_bounds__(512)
void ripple(const int* __restrict__ pos, const int* __restrict__ mh,
            const int* __restrict__ mr, const int* __restrict__ mt,
            const float* __restrict__ iEmb, const float* __restrict__ ent,
            const float* __restrict__ rel, const float* __restrict__ W,
            float* __restrict__ out_i, int B) {
    __shared__ float sW[32 * 32];
    __shared__ float sX[16][32];

    const int lane = threadIdx.x & 31;
    const int wv   = threadIdx.x >> 5;          // 0..15
    const int b    = blockIdx.x * 16 + wv;

    for (int i = threadIdx.x; i < 1024; i += blockDim.x) sW[i] = W[i];

    float itemv = 0.f;
    if (b < B) itemv = iEmb[(size_t)pos[b] * 32 + lane];
    sX[wv][lane] = itemv;
    __syncthreads();

    for (int hop = 0; hop < 2; ++hop) {
        itemv = sX[wv][lane];
        if (b < B) {
            const int base = (hop * B + b) * 32;
            float logit = 0.f;
            for (int m = 0; m < 32; ++m) {
                const int hIdx = mh[base + m];
                const int rIdx = mr[base + m];
                const float hv = ent[(size_t)hIdx * 32 + lane];       // lane e holds h[e]
                const f4* rrow = (const f4*)(rel + (size_t)rIdx * 1024 + (size_t)lane * 32);
                float rh = 0.f;                                        // lane d computes Rh[d]
                #pragma unroll
                for (int j = 0; j < 8; ++j) {
                    f4 rv = rrow[j];
                    rh += rv.x * __shfl(hv, 4 * j + 0, 32);
                    rh += rv.y * __shfl(hv, 4 * j + 1, 32);
                    rh += rv.z * __shfl(hv, 4 * j + 2, 32);
                    rh += rv.w * __shfl(hv, 4 * j + 3, 32);
                }
                float dotv = rh * itemv;                               // dot(Rh, item)
                #pragma unroll
                for (int off = 16; off >= 1; off >>= 1)
                    dotv += __shfl_xor(dotv, off, 32);
                if (lane == m) logit = dotv;                           // lane m keeps logit_m
            }
            // softmax over m (lane m holds logit m)
            float mx = logit;
            #pragma unroll
            for (int off = 16; off >= 1; off >>= 1)
                mx = fmaxf(mx, __shfl_xor(mx, off, 32));
            float e = __expf(logit - mx);
            float se = e;
            #pragma unroll
            for (int off = 16; off >= 1; off >>= 1)
                se += __shfl_xor(se, off, 32);
            const float p = e / se;
            // o[d] = sum_m p_m * leaky(t[m][d])
            float o = 0.f;
            for (int m = 0; m < 32; ++m) {
                const int tIdx = mt[base + m];
                const float tv = leaky1(ent[(size_t)tIdx * 32 + lane]);
                o += __shfl(p, m, 32) * tv;
            }
            sX[wv][lane] = itemv + o;
        }
        __syncthreads();

        // ---- X <- X @ W^T via V_WMMA_F32_16X16X4_F32, wave 0 only (EXEC all 1s) ----
        if (wv == 0) {
            v8f acc0 = {0.f, 0.f, 0.f, 0.f, 0.f, 0.f, 0.f, 0.f};
            v8f acc1 = acc0;
            const int rsel = lane >> 4;     // ISA fp32 A layout: lanes16-31 hold K+2,K+3
            const int l15  = lane & 15;
            #pragma unroll
            for (int k0 = 0; k0 < 32; k0 += 4) {
                const int k = k0 + rsel * 2;
                v2f a;  a.x  = sX[l15][k];               a.y  = sX[l15][k + 1];
                v2f b0; b0.x = sW[l15 * 32 + k];         b0.y = sW[l15 * 32 + k + 1];        // B[k][n]=W[n][k], n=0..15
                v2f b1; b1.x = sW[(l15 + 16) * 32 + k];  b1.y = sW[(l15 + 16) * 32 + k + 1]; // n=16..31
                acc0 = __builtin_amdgcn_wmma_f32_16x16x4_f32(
                    false, a, false, b0, (short)0, acc0, false, false);
                acc1 = __builtin_amdgcn_wmma_f32_16x16x4_f32(
                    false, a, false, b1, (short)0, acc1, false, false);
            }
            // C/D layout: VGPR v -> row v (lanes 0-15) / row v+8 (lanes 16-31)
            #pragma unroll
            for (int vi = 0; vi < 8; ++vi) {
                const int row = vi + rsel * 8;
                sX[row][l15]      = acc0[vi];
                sX[row][l15 + 16] = acc1[vi];
            }
        }
        __syncthreads();
    }

    if (b < B) {
        out_i[(size_t)pos[b] * 32 + lane] = sX[wv][lane];   // scatter updated item rows
    }
}

extern "C" void kernel_launch(void* const* d_in, const int* in_sizes, int n_in,
                              void* d_out, int out_size, void* d_ws, size_t ws_size,
                              hipStream_t stream) {
    const int*   pos = (const int*)d_in[0];
    const int*   mh  = (const int*)d_in[1];
    const int*   mr  = (const int*)d_in[2];
    const int*   mt  = (const int*)d_in[3];
    const float* uE  = (const float*)d_in[4];
    const float* iE  = (const float*)d_in[5];
    const float* ent = (const float*)d_in[6];
    const float* rel = (const float*)d_in[7];
    const float* W   = (const float*)d_in[8];
    float* out = (float*)d_out;

    const int B     = in_sizes[0];   // 2048
    const int slots = in_sizes[1];   // H*B*M = 131072
    const int nU    = in_sizes[4];   // 100000*32
    const int nI    = in_sizes[5];   // 100000*32

    float* out_u = out;
    float* out_i = out_u + nU;
    float* out_h = out_i + nI;
    float* out_t = out_h + (size_t)slots * 32;
    float* out_r = out_t + (size_t)slots * 32;

    const int n4u = nU / 4;
    copy_f4<<<(n4u + 255) / 256, 256, 0, stream>>>((const f4*)uE, (f4*)out_u, n4u);
    const int n4i = nI / 4;
    copy_f4<<<(n4i + 255) / 256, 256, 0, stream>>>((const f4*)iE, (f4*)out_i, n4i);

    const int nht = slots * 8;
    gather_ht<<<(nht + 255) / 256, 256, 0, stream>>>(
        mh, mt, (const f4*)ent, (f4*)out_h, (f4*)out_t, slots);

    gather_r<<<slots, 256, 0, stream>>>(mr, (const f4*)rel, (f4*)out_r);

    ripple<<<(B + 15) / 16, 512, 0, stream>>>(
        pos, mh, mr, mt, iE, ent, rel, W, out_i, B);
}